// ChamferLoss_42494406427162
// MI455X (gfx1250) — compile-verified
//
#include <hip/hip_runtime.h>
#include <math.h>

typedef __attribute__((ext_vector_type(2))) float v2f;
typedef __attribute__((ext_vector_type(4))) float v4f;
typedef __attribute__((ext_vector_type(8))) float v8f;

#define TPB 256
#define WAVES_PER_BLOCK 8
#define ROWS_PER_BLOCK (WAVES_PER_BLOCK * 16)
#define CHUNK_COLS 2048
#define TILES_PER_CHUNK (CHUNK_COLS / 16)   // 128 tiles, 256B apart in LDS

// Preprocess points: out4[i] = {x, y, z, -0.5*||p||^2}
// The .w slot doubles as (a) B-fragment K3 value and (b) -p2/2 for the finish.
__global__ __launch_bounds__(TPB) void chamfer_prep_kernel(
    const float* __restrict__ X, v4f* __restrict__ out4, int n)
{
    const int i = blockIdx.x * TPB + threadIdx.x;
    if (i < n) {
        const float x = X[i * 3 + 0];
        const float y = X[i * 3 + 1];
        const float z = X[i * 3 + 2];
        v4f o;
        o.x = x; o.y = y; o.z = z;
        o.w = -0.5f * __builtin_fmaf(x, x, __builtin_fmaf(y, y, z * z));
        out4[i] = o;
    }
}

// For each row of A, min over all rows of B of squared distance (clamped >= 0).
// A-fragment (16x4): g=0 lanes hold (x,y), g=1 lanes hold (z, 1.0).
// B-fragment (4x16): g=0 lanes hold (x,y), g=1 lanes hold (z, -||t||^2/2).
// WMMA gives c = p.t - t^2/2 ; min sq = p2 - 2*max(c) = -2*(w_p + max(c)).
__global__ __launch_bounds__(TPB) void chamfer_rowmin_kernel(
    const v4f* __restrict__ prepA,
    const v4f* __restrict__ prepB, int nB,
    float* __restrict__ out_minsq)
{
    __shared__ v4f ldsB[CHUNK_COLS];   // 32 KB staged B chunk, fragment-ready

    const int lane = threadIdx.x & 31;
    const int wave = threadIdx.x >> 5;
    const int g    = lane >> 4;        // half-wave 0/1
    const int lm   = lane & 15;
    const int rowBase = (blockIdx.x * WAVES_PER_BLOCK + wave) * 16;

    // A fragment: row = rowBase + lm for both halves.
    const v4f pa = prepA[rowBase + lm];
    v2f a;
    if (g == 0) { a.x = pa.x; a.y = pa.y; }
    else        { a.x = pa.z; a.y = 1.0f; }

    float acc[8];
#pragma unroll
    for (int i = 0; i < 8; ++i) acc[i] = -3.402823466e38f;

    const char* ldsBytes = (const char*)ldsB;
    const int lane_off = lm * 16 + g * 8;   // bank-conflict-free b64 pattern

    for (int cb = 0; cb < nB; cb += CHUNK_COLS) {
        __syncthreads();   // protect previous chunk's readers
        for (int c = threadIdx.x; c < CHUNK_COLS; c += TPB)
            ldsB[c] = prepB[cb + c];          // global_load_b128 + ds_store_b128
        __syncthreads();

        if (cb + CHUNK_COLS < nB)             // warm L2/WGP$ for next chunk
            __builtin_prefetch((const void*)&prepB[cb + CHUNK_COLS + threadIdx.x], 0, 0);

#pragma unroll 4
        for (int t = 0; t < TILES_PER_CHUNK; t += 2) {
            const v2f b0 = *(const v2f*)(ldsBytes + t * 256 + lane_off);
            const v2f b1 = *(const v2f*)(ldsBytes + t * 256 + 256 + lane_off);
            v8f c0 = {};
            v8f c1 = {};
            c0 = __builtin_amdgcn_wmma_f32_16x16x4_f32(false, a, false, b0,
                                                       (short)0, c0, false, false);
            c1 = __builtin_amdgcn_wmma_f32_16x16x4_f32(false, a, false, b1,
                                                       (short)0, c1, false, false);
#pragma unroll
            for (int i = 0; i < 8; ++i)
                acc[i] = fmaxf(acc[i], fmaxf(c0[i], c1[i]));   // -> v_max3_num_f32
        }
    }

    // Reduce across the 16 lanes of each half (xor 1,2,4,8 stays in-half).
#pragma unroll
    for (int i = 0; i < 8; ++i) {
        float v = acc[i];
        v = fmaxf(v, __shfl_xor(v, 1, 32));
        v = fmaxf(v, __shfl_xor(v, 2, 32));
        v = fmaxf(v, __shfl_xor(v, 4, 32));
        v = fmaxf(v, __shfl_xor(v, 8, 32));
        acc[i] = v;
    }

    if (lm == 0) {
        const float* prepAf = (const float*)prepA;
#pragma unroll
        for (int i = 0; i < 8; ++i) {
            const int r = rowBase + i + 8 * g;      // c[i] maps to row i + 8*g
            const float w = prepAf[r * 4 + 3];      // -0.5 * ||p||^2
            const float sq = -2.0f * (w + acc[i]);  // ||p||^2 - 2*max
            out_minsq[r] = sq > 0.0f ? sq : 0.0f;
        }
    }
}

// out[0] = mean(sqrt(minsq_p)) + mean(sqrt(minsq_t)). Single block.
__global__ __launch_bounds__(TPB) void chamfer_reduce_kernel(
    const float* __restrict__ minsq_p, int n,
    const float* __restrict__ minsq_t, int m,
    float* __restrict__ out)
{
    __shared__ float smem[TPB];
    float accP = 0.0f, accT = 0.0f;
    for (int i = threadIdx.x; i < n; i += TPB) accP += sqrtf(minsq_p[i]);
    for (int j = threadIdx.x; j < m; j += TPB) accT += sqrtf(minsq_t[j]);
    smem[threadIdx.x] = accP / (float)n + accT / (float)m;
    __syncthreads();
    for (int off = TPB / 2; off > 0; off >>= 1) {
        if ((int)threadIdx.x < off) smem[threadIdx.x] += smem[threadIdx.x + off];
        __syncthreads();
    }
    if (threadIdx.x == 0) out[0] = smem[0];
}

extern "C" void kernel_launch(void* const* d_in, const int* in_sizes, int n_in,
                              void* d_out, int out_size, void* d_ws, size_t ws_size,
                              hipStream_t stream) {
    (void)n_in; (void)out_size; (void)ws_size;
    const float* pred = (const float*)d_in[0];   // (N, 3) f32
    const float* tgt  = (const float*)d_in[1];   // (M, 3) f32
    const int N = in_sizes[0] / 3;               // 16384
    const int M = in_sizes[1] / 3;               // 8192

    v4f*   prepP = (v4f*)d_ws;                   // N * 16 B
    v4f*   prepT = prepP + N;                    // M * 16 B
    float* minP  = (float*)(prepT + M);          // N * 4 B
    float* minT  = minP + N;                     // M * 4 B

    chamfer_prep_kernel<<<(N + TPB - 1) / TPB, TPB, 0, stream>>>(pred, prepP, N);
    chamfer_prep_kernel<<<(M + TPB - 1) / TPB, TPB, 0, stream>>>(tgt,  prepT, M);

    chamfer_rowmin_kernel<<<N / ROWS_PER_BLOCK, TPB, 0, stream>>>(prepP, prepT, M, minP);
    chamfer_rowmin_kernel<<<M / ROWS_PER_BLOCK, TPB, 0, stream>>>(prepT, prepP, N, minT);

    chamfer_reduce_kernel<<<1, TPB, 0, stream>>>(minP, N, minT, M, (float*)d_out);
}